// EdgeLoss_6854767805020
// MI455X (gfx1250) — compile-verified
//
#include <hip/hip_runtime.h>

// ---------------------------------------------------------------------------
// EdgeLoss for MI455X (gfx1250).
// Bandwidth-bound: 2 streaming passes over 88MB of fp32 logits (~8us at
// 23.3TB/s). Streaming passes use 4-pixel/thread b128 loads (512B per
// wave-load). Segment stats via LDS ds_add_f32 atomics (25KB of the 320KB
// WGP LDS per block). mu = diag(1/max(cnt,1)) x seg_sum is a real fp32
// matmul and is done with V_WMMA_F32_16X16X4_F32.
// ---------------------------------------------------------------------------

#define BATCH   4
#define CH      21
#define HW      (512 * 512)
#define NLAB    256
#define CPAD    32          // global seg/mu row stride (channels padded)
#define LPAD    24          // LDS histogram/mu row stride (96B, 16B aligned)
#define PIXB    4096        // pixels per block in streaming passes
#define BLKS_PER_B (HW / PIXB)   // 64
#define DELTA   0.1f

typedef __attribute__((ext_vector_type(2))) float v2f;
typedef __attribute__((ext_vector_type(4))) float v4f;
typedef __attribute__((ext_vector_type(8))) float v8f;

// ---------------- kernel 0: zero workspace ---------------------------------
__global__ void el_zero_kernel(float* __restrict__ ws, int n) {
    int i = blockIdx.x * blockDim.x + threadIdx.x;
    for (; i < n; i += gridDim.x * blockDim.x) ws[i] = 0.0f;
}

// ---- shared helper: softmax over 21 channels for 4 pixels (b128 loads) ----
__device__ __forceinline__ void softmax4(const float* __restrict__ pb, int n,
                                         v4f (&x)[CH], v4f& inv) {
    v4f mx = {-1e30f, -1e30f, -1e30f, -1e30f};
#pragma unroll
    for (int c = 0; c < CH; ++c) {
        x[c] = *(const v4f*)(pb + (size_t)c * HW + n);   // global_load_b128
#pragma unroll
        for (int k = 0; k < 4; ++k) mx[k] = fmaxf(mx[k], x[c][k]);
    }
    v4f s = {0.f, 0.f, 0.f, 0.f};
#pragma unroll
    for (int c = 0; c < CH; ++c) {
#pragma unroll
        for (int k = 0; k < 4; ++k) {
            x[c][k] = __expf(x[c][k] - mx[k]);
            s[k] += x[c][k];
        }
    }
#pragma unroll
    for (int k = 0; k < 4; ++k) inv[k] = 1.0f / s[k];
}

// ---------------- kernel 1: softmax + per-label segment sums ---------------
__global__ void __launch_bounds__(256)
el_seg_kernel(const float* __restrict__ pred, const long long* __restrict__ edge,
              float* __restrict__ seg, float* __restrict__ cnt) {
    __shared__ float lseg[NLAB * LPAD];
    __shared__ float lcnt[NLAB];

    const int b    = blockIdx.x / BLKS_PER_B;
    const int base = (blockIdx.x % BLKS_PER_B) * PIXB;
    const float* pb = pred + (size_t)b * CH * HW;
    const long long* eb = edge + (size_t)b * HW;

    for (int i = threadIdx.x; i < NLAB * LPAD; i += 256) lseg[i] = 0.0f;
    for (int i = threadIdx.x; i < NLAB; i += 256) lcnt[i] = 0.0f;
    __syncthreads();

    for (int it = 0; it < PIXB / 1024; ++it) {           // 4 iters x 1024 px
        const int n = base + it * 1024 + threadIdx.x * 4;
        __builtin_prefetch(pb + n + 1024, 0, 0);         // global_prefetch_b8

        v4f x[CH], inv;
        softmax4(pb, n, x, inv);

        const long long* ep = eb + n;
#pragma unroll
        for (int k = 0; k < 4; ++k) {
            const int e = (int)ep[k];
            const float invk = inv[k];
#pragma unroll
            for (int c = 0; c < CH; ++c)
                atomicAdd(&lseg[e * LPAD + c], x[c][k] * invk);  // ds_add_f32
            atomicAdd(&lcnt[e], 1.0f);
        }
    }
    __syncthreads();

    // flush block histogram to global (f32 HW atomics, L2-resident table)
    for (int i = threadIdx.x; i < NLAB * LPAD; i += 256) {
        const int l = i / LPAD, c = i % LPAD;
        if (c < CH) {
            const float v = lseg[i];
            if (v != 0.0f)
                unsafeAtomicAdd(&seg[((size_t)b * NLAB + l) * CPAD + c], v);
        }
    }
    for (int i = threadIdx.x; i < NLAB; i += 256) {
        const float v = lcnt[i];
        if (v != 0.0f) unsafeAtomicAdd(&cnt[b * NLAB + i], v);
    }
}

// ---------------- kernel 2: mu = diag(1/max(cnt,1)) x seg  (WMMA f32) ------
// One wave per 16-label x 16-channel tile. D[16x16] accumulates 4 K-chunks of
// V_WMMA_F32_16X16X4_F32 with a diagonal A fragment built in registers.
// A 16x4 f32 layout: lane L holds row M=L&15; VGPR0 holds K = (L<16?0:2),
// VGPR1 holds K = (L<16?1:3).  B 4x16 layout mirrors C/D row striping.
__global__ void __launch_bounds__(256)
el_mu_wmma_kernel(const float* __restrict__ seg, const float* __restrict__ cnt,
                  float* __restrict__ mu) {
    const int wid  = (blockIdx.x * 256 + threadIdx.x) >> 5;  // 0..127
    const int lane = threadIdx.x & 31;
    const int b  = wid >> 5;          // 0..3
    const int r  = wid & 31;
    const int lb = r >> 1;            // 16-label block 0..15
    const int cbase = (r & 1) * 16;   // channel tile base 0 / 16

    const int l15 = lane & 15;
    const int hi  = (lane >> 4) & 1;

    const float inv = 1.0f / fmaxf(cnt[b * NLAB + lb * 16 + l15], 1.0f);

    v8f acc = {0.f, 0.f, 0.f, 0.f, 0.f, 0.f, 0.f, 0.f};
    const float* sp = seg + ((size_t)b * NLAB + lb * 16) * CPAD + cbase + l15;

#pragma unroll
    for (int c = 0; c < 4; ++c) {
        const int kk0 = c * 4 + hi * 2;     // K index held in VGPR0 slot
        v2f a, bm;
        a.x = (l15 == kk0)     ? inv : 0.0f;
        a.y = (l15 == kk0 + 1) ? inv : 0.0f;
        bm.x = sp[(size_t)kk0 * CPAD];
        bm.y = sp[(size_t)(kk0 + 1) * CPAD];
        acc = __builtin_amdgcn_wmma_f32_16x16x4_f32(
            /*neg_a=*/false, a, /*neg_b=*/false, bm,
            /*c_mod=*/(short)0, acc, /*reuse_a=*/false, /*reuse_b=*/false);
    }

    float* mp = mu + ((size_t)b * NLAB + lb * 16 + hi * 8) * CPAD + cbase + l15;
#pragma unroll
    for (int j = 0; j < 8; ++j) mp[(size_t)j * CPAD] = acc[j];
}

// ---------------- kernel 3: distance pass + masked reduction ---------------
__global__ void __launch_bounds__(256)
el_dist_kernel(const float* __restrict__ pred, const long long* __restrict__ edge,
               const float* __restrict__ mu, float* __restrict__ numden) {
    __shared__ float lmu[NLAB * LPAD];
    __shared__ float snum, sden;

    const int b    = blockIdx.x / BLKS_PER_B;
    const int base = (blockIdx.x % BLKS_PER_B) * PIXB;
    const float* pb = pred + (size_t)b * CH * HW;
    const long long* eb = edge + (size_t)b * HW;

    if (threadIdx.x == 0) { snum = 0.0f; sden = 0.0f; }
    // stage this sample's mu table (24KB) into LDS; cols 21..23 are zero
    for (int i = threadIdx.x; i < NLAB * LPAD; i += 256)
        lmu[i] = mu[((size_t)b * NLAB + i / LPAD) * CPAD + (i % LPAD)];
    __syncthreads();

    float ln = 0.0f, ld = 0.0f;
    for (int it = 0; it < PIXB / 1024; ++it) {
        const int n = base + it * 1024 + threadIdx.x * 4;
        __builtin_prefetch(pb + n + 1024, 0, 0);

        v4f x[CH], inv;
        softmax4(pb, n, x, inv);

        const long long* ep = eb + n;
#pragma unroll
        for (int k = 0; k < 4; ++k) {
            const int e = (int)ep[k];
            const float invk = inv[k];
            // mu row: 96B-aligned LDS row -> 5x ds_load_b128 + 1x b32
            const v4f* mrow4 = (const v4f*)&lmu[e * LPAD];
            v4f mr[5];
#pragma unroll
            for (int q = 0; q < 5; ++q) mr[q] = mrow4[q];
            const float mlast = lmu[e * LPAD + 20];

            float d = 0.0f;
#pragma unroll
            for (int c = 0; c < 20; ++c)
                d += fabsf(x[c][k] * invk - mr[c >> 2][c & 3]);
            d += fabsf(x[20][k] * invk - mlast);
            d = fmaxf(d - DELTA, 0.0f);

            if ((e != 0) && (e != 255)) { ln += d; ld += 1.0f; }
        }
    }

    atomicAdd(&snum, ln);
    atomicAdd(&sden, ld);
    __syncthreads();
    if (threadIdx.x == 0) {
        unsafeAtomicAdd(&numden[b], snum);
        unsafeAtomicAdd(&numden[BATCH + b], sden);
    }
}

// ---------------- kernel 4: finalize ---------------------------------------
__global__ void el_final_kernel(const float* __restrict__ numden,
                                float* __restrict__ out) {
    if (threadIdx.x == 0 && blockIdx.x == 0) {
        float acc = 0.0f;
        for (int b = 0; b < BATCH; ++b)
            acc += numden[b] / (numden[BATCH + b] + 1e-5f);
        out[0] = acc / (float)BATCH;
    }
}

// ---------------------------------------------------------------------------
extern "C" void kernel_launch(void* const* d_in, const int* in_sizes, int n_in,
                              void* d_out, int out_size, void* d_ws, size_t ws_size,
                              hipStream_t stream) {
    const float*     pred = (const float*)d_in[0];      // [4,21,512,512] f32
    const long long* edge = (const long long*)d_in[1];  // [4,512,512]  i64

    float* ws     = (float*)d_ws;
    float* seg    = ws;                               // B*256*32
    float* cnt    = seg + BATCH * NLAB * CPAD;        // B*256
    float* mu     = cnt + BATCH * NLAB;               // B*256*32
    float* numden = mu + BATCH * NLAB * CPAD;         // 2*B

    const int zn = BATCH * NLAB * CPAD * 2 + BATCH * NLAB + 2 * BATCH;
    el_zero_kernel<<<64, 256, 0, stream>>>(ws, zn);

    el_seg_kernel<<<BATCH * BLKS_PER_B, 256, 0, stream>>>(pred, edge, seg, cnt);
    el_mu_wmma_kernel<<<16, 256, 0, stream>>>(seg, cnt, mu);
    el_dist_kernel<<<BATCH * BLKS_PER_B, 256, 0, stream>>>(pred, edge, mu, numden);
    el_final_kernel<<<1, 64, 0, stream>>>(numden, (float*)d_out);
}